// ModuleCollect_76647986364629
// MI455X (gfx1250) — compile-verified
//
#include <hip/hip_runtime.h>
#include <stdint.h>

// Gather+concat: out[n] = [x[n], m0*x[i0], m1*x[i1], m2*x[i2], m3*x[i3]]
// N=100000, D=256 fp32. Pure data movement (0 FLOPs, ~615MB HBM traffic at the
// roofline -> ~26us @ 23.3TB/s). CDNA5 path: async global<->LDS DMA (ASYNCcnt)
// with TH_STORE_NT output streaming so x (102MB) stays resident in the 192MB L2
// and all gather reads are L2 hits.

typedef float v4f __attribute__((ext_vector_type(4)));
typedef int   v4i __attribute__((ext_vector_type(4)));

#define D_IN 256
constexpr int kRowBytes      = D_IN * 4;        // 1024 B per x row
constexpr int kOutRowBytes   = 5 * kRowBytes;   // 5120 B per out row
constexpr int kWavesPerBlock = 8;
constexpr int kBlock         = 256;             // 8 wave32

__global__ __launch_bounds__(kBlock)
void gather_concat_async(const float* __restrict__ x,
                         const int* __restrict__ indices,
                         float* __restrict__ out,
                         int n)
{
    // one private 5KB slab per wave: laid out exactly like the output row
    __shared__ __align__(16) unsigned char smem[kWavesPerBlock * kOutRowBytes];

    const unsigned lane = threadIdx.x & 31u;
    const unsigned wave = threadIdx.x >> 5;
    // row is uniform per wave: force it scalar so the guard and all base
    // addressing is SALU (s_cmp/s_cbranch instead of exec masking)
    const int row = __builtin_amdgcn_readfirstlane(
        (int)(blockIdx.x * kWavesPerBlock + wave));
    if (row >= n) return;

    unsigned char* wbuf = &smem[wave * (unsigned)kOutRowBytes];
    // low 32 bits of a generic pointer to LDS == wave-relative LDS byte offset
    const unsigned lds_lane = (unsigned)(uintptr_t)wbuf + lane * 16u;

    // ---- kick off the neighbor-index fetch first (single b128, one wait
    //      that overlaps with the seg-0 DMA issue below)
    const v4i nbv = *(const v4i*)(indices + (size_t)(unsigned)row * 4u);

    // ---- segment 0: x[row] -> the first 1024 bytes of the wave's LDS slab.  2 x b128 = 2 x 512B per wave.
    {
        uint64_t src = (uint64_t)(uintptr_t)x
                     + (uint64_t)(unsigned)row * (unsigned)kRowBytes + lane * 16u;
        asm volatile("global_load_async_to_lds_b128 %0, %1, off"
                     :: "v"(lds_lane), "v"(src) : "memory");
        asm volatile("global_load_async_to_lds_b128 %0, %1, off offset:512"
                     :: "v"(lds_lane), "v"(src) : "memory");
    }

    // ---- neighbor segments: wave-uniform indices -> scalar branches
    const int nb_arr[4] = {
        __builtin_amdgcn_readfirstlane(nbv.x),
        __builtin_amdgcn_readfirstlane(nbv.y),
        __builtin_amdgcn_readfirstlane(nbv.z),
        __builtin_amdgcn_readfirstlane(nbv.w)
    };
    const v4f zero = {0.f, 0.f, 0.f, 0.f};
#pragma unroll
    for (int j = 0; j < 4; ++j) {
        const int nb = nb_arr[j];
        const unsigned seg_lane = lds_lane + (unsigned)(j + 1) * (unsigned)kRowBytes;
        if (nb >= 0) {
            uint64_t src = (uint64_t)(uintptr_t)x
                         + (uint64_t)(unsigned)nb * (unsigned)kRowBytes + lane * 16u;
            asm volatile("global_load_async_to_lds_b128 %0, %1, off"
                         :: "v"(seg_lane), "v"(src) : "memory");
            asm volatile("global_load_async_to_lds_b128 %0, %1, off offset:512"
                         :: "v"(seg_lane), "v"(src) : "memory");
        } else {
            // masked neighbor: zero-fill the LDS segment (ds_store_b128 x2)
            v4f* p = (v4f*)(wbuf + (unsigned)(j + 1) * (unsigned)kRowBytes) + lane;
            p[0]  = zero;   // bytes [lane*16, lane*16+16)
            p[32] = zero;   // bytes [512 + lane*16, ...)
        }
    }

    // all async loads landed in LDS; all ds zero-fills done
    asm volatile("s_wait_asynccnt 0x0" ::: "memory");
    asm volatile("s_wait_dscnt 0x0" ::: "memory");

    // ---- stream the contiguous 5KB output row: 10 x b128 async stores,
    //      non-temporal so the 512MB output doesn't evict x from L2.
    uint64_t dst = (uint64_t)(uintptr_t)out
                 + (uint64_t)(unsigned)row * (unsigned)kOutRowBytes + lane * 16u;
#pragma unroll
    for (int k = 0; k < 10; ++k) {
        asm volatile("global_store_async_from_lds_b128 %0, %1, off offset:%c2 th:TH_STORE_NT"
                     :: "v"(dst), "v"(lds_lane), "i"(k * 512) : "memory");
    }
    asm volatile("s_wait_asynccnt 0x0" ::: "memory");
}

extern "C" void kernel_launch(void* const* d_in, const int* in_sizes, int n_in,
                              void* d_out, int out_size, void* d_ws, size_t ws_size,
                              hipStream_t stream) {
    const float* x   = (const float*)d_in[0];
    const int* idx   = (const int*)d_in[1];
    float* out       = (float*)d_out;
    (void)n_in; (void)out_size; (void)d_ws; (void)ws_size;

    const int n = in_sizes[0] / D_IN;                       // 100000 rows
    const int blocks = (n + kWavesPerBlock - 1) / kWavesPerBlock;
    gather_concat_async<<<blocks, kBlock, 0, stream>>>(x, idx, out, n);
}